// FSQ_80599356277526
// MI455X (gfx1250) — compile-verified
//
#include <hip/hip_runtime.h>

// FSQ forward for MI455X (gfx1250, wave32).
// Streaming problem: 32MB in, 40MB out -> ~3.1us HBM-bound at 23.3 TB/s.
// No matrix ops exist in this workload (no WMMA). CDNA5 paths used:
// b128 NT loads/stores (th:NT), global_prefetch_b8, v_exp_f32/v_rcp_f32 TRANS.
//
// Key change vs previous round: ocml tanhf cost ~36 VALU/element (valu=578 in
// the loop) which made the kernel VALU-bound. Outputs only consume tanh via a
// nearest-level threshold, so we use (tanh(x)+1)*(n-1)/2 == (n-1)*e/(e+1),
// e = exp2(2*log2e*x), with hardware exp2 + rcp (+1 Newton step): ~16 VALU/elem.

typedef float v4f __attribute__((ext_vector_type(4)));
typedef int   v4i __attribute__((ext_vector_type(4)));

// Nearest level of lv = linspace(-1,1,n) for tanh(xin):
//   t   = (tanh(xin)+1)*(n-1)/2 = (n-1)*e/(e+1),  e = exp2(x * 2*log2(e))
//   idx = clamp(rint(t), 0, n-1);  q = idx*(2/(n-1)) - 1
__device__ __forceinline__ void fsq_quant(float xin, float nm1f, float step,
                                          float& q, int& idx) {
    float a = fminf(fmaxf(__fmul_rn(xin, 2.885390081777927f), -64.0f), 64.0f);
    float e = __builtin_amdgcn_exp2f(a);          // v_exp_f32 (TRANS)
    float d = __fadd_rn(e, 1.0f);
    float r = __builtin_amdgcn_rcpf(d);           // v_rcp_f32 (TRANS)
    r = __fmul_rn(r, __builtin_fmaf(-d, r, 2.0f));// Newton: ~0.5 ulp rcp
    float t = __fmul_rn(__fmul_rn(nm1f, e), r);   // in [0, n-1]
    float f = fminf(fmaxf(rintf(t), 0.0f), nm1f); // v_rndne + clamp
    idx = (int)f;
    // match linspace's k*step + start evaluation (separate mul, add; no fma)
    q = __fadd_rn(__fmul_rn(f, step), -1.0f);
}

__global__ void __launch_bounds__(256)
fsq_fwd_kernel(const float* __restrict__ x,    // (N,4) rows
               float*       __restrict__ z,    // (4,N) transposed streams
               int*         __restrict__ code, // (N,)
               int nrows) {                    // N, multiple of 4
    const int nchunks = nrows >> 2;            // 4 rows per thread-chunk
    const int stride  = (int)(gridDim.x * blockDim.x);

    float* __restrict__ z0 = z;
    float* __restrict__ z1 = z + (size_t)nrows;
    float* __restrict__ z2 = z + 2u * (size_t)nrows;
    float* __restrict__ z3 = z + 3u * (size_t)nrows;

    for (int c = (int)(blockIdx.x * blockDim.x + threadIdx.x); c < nchunks; c += stride) {
        const v4f* __restrict__ xin = (const v4f*)x + (size_t)c * 4u;

        // gfx1250 speculative prefetch of next grid-stride tile (guarded).
        if (c + stride < nchunks)
            __builtin_prefetch((const void*)((const v4f*)x + ((size_t)c + (size_t)stride) * 4u), 0, 0);

        v4f qs0, qs1, qs2, qs3;   // per-level output streams (register transpose)
        v4i cd;                   // packed codes for 4 rows

#pragma unroll
        for (int r = 0; r < 4; ++r) {
            v4f v = __builtin_nontemporal_load(xin + r);   // global_load_b128 th:NT
            float q0, q1, q2, q3;
            int   i0, i1, i2, i3;
            fsq_quant(v.x, 7.0f, 2.0f / 7.0f, q0, i0);     // n=8
            fsq_quant(v.y, 4.0f, 0.5f,        q1, i1);     // n=5
            fsq_quant(v.z, 4.0f, 0.5f,        q2, i2);     // n=5
            fsq_quant(v.w, 4.0f, 0.5f,        q3, i3);     // n=5
            qs0[r] = q0; qs1[r] = q1; qs2[r] = q2; qs3[r] = q3;
            cd[r]  = i0 + 8 * i1 + 40 * i2 + 200 * i3;     // mult = {1,8,40,200}
        }

        // Five fully-coalesced b128 NT stores per thread.
        __builtin_nontemporal_store(qs0, (v4f*)z0 + c);
        __builtin_nontemporal_store(qs1, (v4f*)z1 + c);
        __builtin_nontemporal_store(qs2, (v4f*)z2 + c);
        __builtin_nontemporal_store(qs3, (v4f*)z3 + c);
        __builtin_nontemporal_store(cd,  (v4i*)code + c);
    }
}

extern "C" void kernel_launch(void* const* d_in, const int* in_sizes, int n_in,
                              void* d_out, int out_size, void* d_ws, size_t ws_size,
                              hipStream_t stream) {
    (void)n_in; (void)out_size; (void)d_ws; (void)ws_size;

    const float* x = (const float*)d_in[0];
    const int total = in_sizes[0];        // B*T*D = 8,388,608
    const int nrows = total / 4;          // B*T   = 2,097,152

    float* z    = (float*)d_out;                    // 4*nrows floats (level-major)
    int*   code = (int*)((float*)d_out + total);    // nrows int32 codes

    // 1024 blocks x 256 threads (8 wave32 waves/block): 262,144 threads,
    // exactly 2 grid-stride iterations over the 524,288 chunks -> prefetch
    // covers iteration 2 while HBM stays saturated.
    const int block = 256;
    const int grid  = 1024;
    fsq_fwd_kernel<<<grid, block, 0, stream>>>(x, z, code, nrows);
}